// Encoder_64020782514980
// MI455X (gfx1250) — compile-verified
//
#include <hip/hip_runtime.h>

#define NN 100000
#define EE 1600000
#define FF 128
#define HH 128
#define GG 64

typedef float v2f __attribute__((ext_vector_type(2)));
typedef float v8f __attribute__((ext_vector_type(8)));

// ---------------- utility kernels ----------------

__global__ void fill_f32(float* __restrict__ p, float v, int n) {
  int i = blockIdx.x * blockDim.x + threadIdx.x;
  if (i < n) p[i] = v;
}

__global__ void fill_i32(int* __restrict__ p, int v, int n) {
  int i = blockIdx.x * blockDim.x + threadIdx.x;
  if (i < n) p[i] = v;
}

// count in-degree (dst-based) with int atomics
__global__ void deg_count(const int* __restrict__ dst, int* __restrict__ cnt, int ne) {
  int e = blockIdx.x * blockDim.x + threadIdx.x;
  if (e < ne) atomicAdd(cnt + dst[e], 1);
}

// dis[i] = rsqrt(deg[i] + 1)   (self-loop included)
__global__ void make_dis(const int* __restrict__ cnt, float* __restrict__ dis, int n) {
  int i = blockIdx.x * blockDim.x + threadIdx.x;
  if (i < n) dis[i] = rsqrtf((float)cnt[i] + 1.0f);
}

// single-block exclusive scan: rowptr[0..N] from cnt[0..N-1]
__global__ void scan_rowptr(const int* __restrict__ cnt, int* __restrict__ rowptr) {
  __shared__ int ssum[1024];
  __shared__ int soff[1024];
  const int t = threadIdx.x;
  const int CH = (NN + 1023) / 1024;
  int lo = t * CH; if (lo > NN) lo = NN;
  int hi = lo + CH; if (hi > NN) hi = NN;
  int s = 0;
  for (int i = lo; i < hi; ++i) s += cnt[i];
  ssum[t] = s;
  __syncthreads();
  if (t == 0) {
    int acc = 0;
    for (int i = 0; i < 1024; ++i) { soff[i] = acc; acc += ssum[i]; }
  }
  __syncthreads();
  int acc = soff[t];
  for (int i = lo; i < hi; ++i) { rowptr[i] = acc; acc += cnt[i]; }
  if (t == 1023) rowptr[NN] = acc;   // == EE
}

// bucket-fill edge ids by dst
__global__ void csr_fill(const int* __restrict__ dst, const int* __restrict__ rowptr,
                         int* __restrict__ cur, int* __restrict__ eidx, int ne) {
  int e = blockIdx.x * blockDim.x + threadIdx.x;
  if (e >= ne) return;
  int d = dst[e];
  int p = atomicAdd(cur + d, 1);
  eidx[rowptr[d] + p] = e;
}

// ---------------- WMMA fp32 GEMM: Out[M,OUTC] = A[M,K] @ W[K,OUTC] ----------------
// One wave computes MT vertically-stacked 16x16 tiles (B fragment reused MT times).
// block = 256 threads = 8 waves; grid = (M/(16*MT), OUTC/128). K, OUTC compile-time.
// Lane layout (ISA 7.12.2):
//   A (16x4):  lane L holds row M = L%16; VGPR0/1 hold K = 2*(L/16) + {0,1}
//   B (4x16):  lane L holds col N = L%16; VGPR0/1 hold K = 2*(L/16) + {0,1}
//   C (16x16): VGPR r holds row r + 8*(L/16), col L%16
template <int MT, int K, int OUTC>
__global__ void gemm_wmma(const float* __restrict__ A, const float* __restrict__ W,
                          float* __restrict__ Out) {
  const int lane = threadIdx.x & 31;
  const int wave = threadIdx.x >> 5;
  const int l16  = lane & 15;
  const int half = lane >> 4;                 // 0 or 1
  const int row0 = blockIdx.x * (16 * MT);
  const int col0 = (blockIdx.y * 8 + wave) * 16;

  const float* Ap = A + (size_t)(row0 + l16) * K + half * 2;
  const float* Wp = W + (size_t)(half * 2) * OUTC + col0 + l16;

  v8f acc[MT];
#pragma unroll
  for (int m = 0; m < MT; ++m) acc[m] = (v8f){};

#pragma unroll 8
  for (int k = 0; k < K; k += 4) {
    v2f b;
    b.x = Wp[(size_t)k * OUTC];
    b.y = Wp[(size_t)(k + 1) * OUTC];
#pragma unroll
    for (int m = 0; m < MT; ++m) {
      const float* ap = Ap + (size_t)(m * 16) * K + k;
      v2f a; a.x = ap[0]; a.y = ap[1];
      acc[m] = __builtin_amdgcn_wmma_f32_16x16x4_f32(false, a, false, b, (short)0,
                                                     acc[m], false, false);
    }
  }

#pragma unroll
  for (int m = 0; m < MT; ++m) {
    float* Op = Out + (size_t)(row0 + m * 16 + half * 8) * OUTC + col0 + l16;
#pragma unroll
    for (int r = 0; r < 8; ++r) Op[(size_t)r * OUTC] = acc[m][r];
  }
}

// ---------------- gather aggregation (CSR by dst, no float atomics) ----------------
// one wave per node; lanes stripe channels in float4. OUTC compile-time (128 or 256).
// X[i] = act( b + H[i]*dis[i]^2 + sum_{e: dst=i} H[src_e]*dis[src_e]*dis[i] )
template <bool RELU, int OUTC>
__global__ void gcn_aggregate(const int* __restrict__ rowptr, const int* __restrict__ eidx,
                              const int* __restrict__ src, const float* __restrict__ dis,
                              const float* __restrict__ Hh, const float* __restrict__ bias,
                              float* __restrict__ X) {
  int i = blockIdx.x * (blockDim.x >> 5) + (threadIdx.x >> 5);
  if (i >= NN) return;
  const int lane = threadIdx.x & 31;
  constexpr int NV = OUTC / 128;              // float4s per lane (1 or 2)

  const float di = dis[i];
  const float d2 = di * di;

  const float4* hi4 = (const float4*)(Hh + (size_t)i * OUTC);
  const float4* b4  = (const float4*)bias;

  float4 acc[NV];
#pragma unroll
  for (int v = 0; v < NV; ++v) {
    float4 h = hi4[lane + 32 * v], b = b4[lane + 32 * v];
    acc[v].x = fmaf(h.x, d2, b.x); acc[v].y = fmaf(h.y, d2, b.y);
    acc[v].z = fmaf(h.z, d2, b.z); acc[v].w = fmaf(h.w, d2, b.w);
  }

  const int p0 = rowptr[i], p1 = rowptr[i + 1];
  for (int p = p0; p < p1; ++p) {
    int e = eidx[p];
    int s = src[e];
    float nrm = dis[s] * di;                  // uniform across wave (broadcast load)
    const float4* hs = (const float4*)(Hh + (size_t)s * OUTC);
#pragma unroll
    for (int v = 0; v < NV; ++v) {
      float4 w = hs[lane + 32 * v];
      acc[v].x = fmaf(w.x, nrm, acc[v].x); acc[v].y = fmaf(w.y, nrm, acc[v].y);
      acc[v].z = fmaf(w.z, nrm, acc[v].z); acc[v].w = fmaf(w.w, nrm, acc[v].w);
    }
  }

  float4* xo = (float4*)(X + (size_t)i * OUTC);
#pragma unroll
  for (int v = 0; v < NV; ++v) {
    float4 r = acc[v];
    if (RELU) {
      r.x = fmaxf(r.x, 0.f); r.y = fmaxf(r.y, 0.f);
      r.z = fmaxf(r.z, 0.f); r.w = fmaxf(r.w, 0.f);
    }
    xo[lane + 32 * v] = r;
  }
}

// ---------------- pooling ----------------

__global__ void pool_count(const int* __restrict__ batch, float* __restrict__ cntg) {
  int i = blockIdx.x * blockDim.x + threadIdx.x;
  if (i < NN) atomicAdd(cntg + batch[i], 1.0f);
}

// sums[batch[i]] += x4[i] + skip[i] + bl
__global__ void pool_sum(const float* __restrict__ X4, const float* __restrict__ Sk,
                         const float* __restrict__ bl, const int* __restrict__ batch,
                         float* __restrict__ sums) {
  int t = blockIdx.x * blockDim.x + threadIdx.x;   // NN * (HH/4) threads
  int i = t >> 5;                                  // HH/4 == 32
  int c = t & 31;
  if (i >= NN) return;
  int g = batch[i];
  float4 a = ((const float4*)(X4 + (size_t)i * HH))[c];
  float4 s = ((const float4*)(Sk + (size_t)i * HH))[c];
  float4 b = ((const float4*)bl)[c];
  float* sp = sums + (size_t)g * HH + c * 4;
  atomicAdd(sp + 0, a.x + s.x + b.x);
  atomicAdd(sp + 1, a.y + s.y + b.y);
  atomicAdd(sp + 2, a.z + s.z + b.z);
  atomicAdd(sp + 3, a.w + s.w + b.w);
}

__global__ void pool_final(const float* __restrict__ sums, const float* __restrict__ cntg,
                           float* __restrict__ out) {
  int t = blockIdx.x * blockDim.x + threadIdx.x;
  if (t >= GG * HH) return;
  int g = t / HH;
  out[t] = sums[t] / fmaxf(cntg[g], 1.0f);
}

// ---------------- driver ----------------

static inline size_t align4(size_t x) { return (x + 3) & ~(size_t)3; }

extern "C" void kernel_launch(void* const* d_in, const int* in_sizes, int n_in,
                              void* d_out, int out_size, void* d_ws, size_t ws_size,
                              hipStream_t stream) {
  const float* x    = (const float*)d_in[0];
  const int*   ei   = (const int*)d_in[1];
  const int*   batch= (const int*)d_in[2];
  const float* W1 = (const float*)d_in[3];
  const float* b1 = (const float*)d_in[4];
  const float* W2 = (const float*)d_in[5];
  const float* b2 = (const float*)d_in[6];
  const float* W3 = (const float*)d_in[7];
  const float* b3 = (const float*)d_in[8];
  const float* W4 = (const float*)d_in[9];
  const float* b4 = (const float*)d_in[10];
  const float* Wl = (const float*)d_in[11];
  const float* bl = (const float*)d_in[12];
  float* out = (float*)d_out;

  const int* src = ei;        // edge_index[0]
  const int* dst = ei + EE;   // edge_index[1]

  // workspace layout (4-byte units, float4 regions kept 16B aligned)
  char* wsb = (char*)d_ws;
  size_t o = 0;
  float* dis   = (float*)(wsb + o * 4); o = align4(o + NN);
  int*   cnt   = (int*)  (wsb + o * 4); o = align4(o + NN);
  int*   cur   = (int*)  (wsb + o * 4); o = align4(o + NN);
  int*   rowp  = (int*)  (wsb + o * 4); o = align4(o + NN + 1);
  int*   eidx  = (int*)  (wsb + o * 4); o = align4(o + EE);
  float* h     = (float*)(wsb + o * 4); o = align4(o + (size_t)NN * 256);
  float* xa    = (float*)(wsb + o * 4); o = align4(o + (size_t)NN * 256);
  float* xb    = (float*)(wsb + o * 4); o = align4(o + (size_t)NN * 256);
  float* skip  = (float*)(wsb + o * 4); o = align4(o + (size_t)NN * 128);
  float* sums  = (float*)(wsb + o * 4); o = align4(o + GG * HH);
  float* cntg  = (float*)(wsb + o * 4); o = align4(o + GG);

  dim3 blk(256);

  // ---- build degree + CSR (by dst) ----
  fill_i32<<<(NN + 255) / 256, blk, 0, stream>>>(cnt, 0, NN);
  deg_count<<<(EE + 255) / 256, blk, 0, stream>>>(dst, cnt, EE);
  make_dis<<<(NN + 255) / 256, blk, 0, stream>>>(cnt, dis, NN);
  scan_rowptr<<<1, 1024, 0, stream>>>(cnt, rowp);
  fill_i32<<<(NN + 255) / 256, blk, 0, stream>>>(cur, 0, NN);
  csr_fill<<<(EE + 255) / 256, blk, 0, stream>>>(dst, rowp, cur, eidx, EE);

  // ---- skip = x @ Wl (bl folded into pooling) ----
  gemm_wmma<5, FF, HH><<<dim3(NN / 80, 1), blk, 0, stream>>>(x, Wl, skip);

  // ---- layer 1: x -> xa ----
  gemm_wmma<5, FF, HH><<<dim3(NN / 80, 1), blk, 0, stream>>>(x, W1, h);
  gcn_aggregate<true, HH><<<(NN + 7) / 8, blk, 0, stream>>>(rowp, eidx, src, dis, h, b1, xa);

  // ---- layer 2: xa -> xb ----
  gemm_wmma<5, HH, HH><<<dim3(NN / 80, 1), blk, 0, stream>>>(xa, W2, h);
  gcn_aggregate<true, HH><<<(NN + 7) / 8, blk, 0, stream>>>(rowp, eidx, src, dis, h, b2, xb);

  // ---- layer 3: xb -> xa (OUT = 256) ----
  gemm_wmma<5, HH, 2 * HH><<<dim3(NN / 80, 2), blk, 0, stream>>>(xb, W3, h);
  gcn_aggregate<true, 2 * HH><<<(NN + 7) / 8, blk, 0, stream>>>(rowp, eidx, src, dis, h, b3, xa);

  // ---- layer 4: xa -> xb (K = 256, no relu) ----
  gemm_wmma<5, 2 * HH, HH><<<dim3(NN / 80, 1), blk, 0, stream>>>(xa, W4, h);
  gcn_aggregate<false, HH><<<(NN + 7) / 8, blk, 0, stream>>>(rowp, eidx, src, dis, h, b4, xb);

  // ---- global mean pool of (xb + skip + bl) ----
  fill_f32<<<(GG * HH + 255) / 256, blk, 0, stream>>>(sums, 0.0f, GG * HH);
  fill_f32<<<1, 64, 0, stream>>>(cntg, 0.0f, GG);
  pool_count<<<(NN + 255) / 256, blk, 0, stream>>>(batch, cntg);
  pool_sum<<<(NN * (HH / 4) + 255) / 256, blk, 0, stream>>>(xb, skip, bl, batch, sums);
  pool_final<<<(GG * HH + 255) / 256, blk, 0, stream>>>(sums, cntg, out);
}